// ModelNew_51659866636814
// MI455X (gfx1250) — compile-verified
//
#include <hip/hip_runtime.h>

typedef __attribute__((ext_vector_type(2))) float v2f;
typedef __attribute__((ext_vector_type(8))) float v8f;
typedef __attribute__((ext_vector_type(4))) unsigned int v4u;
typedef __attribute__((ext_vector_type(4))) int v4i;
typedef __attribute__((ext_vector_type(8))) int v8i;

#define CHUNK    256
#define DDIM     64
#define KSPLIT   4
#define SLABS    16            // d-values per split
#define QSTRIDE  68            // Q LDS row stride (dwords): 64 + 4 pad (TDM pad mode)
#define SSTRIDE  68            // S LDS *column* stride (dwords): transposed slab storage
#define QS_DW    (CHUNK * QSTRIDE)
#define SB_DW    (DDIM * SSTRIDE)   // one slab buffer: 64 columns x 68 dwords

// HBM -> LDS async DMA (CDNA5, tracked by ASYNCcnt)
__device__ __forceinline__ void async_b32_to_lds(unsigned lds_byte_off, const float* gaddr) {
    asm volatile("global_load_async_to_lds_b32 %0, %1, off"
                 :: "v"(lds_byte_off), "v"(gaddr) : "memory");
}
__device__ __forceinline__ void wait_asynccnt0() {
    asm volatile("s_wait_asynccnt 0" ::: "memory");
}

// TDM: one descriptor moves the whole padded Q tile HBM->LDS (tracked by TENSORcnt).
// D# per CDNA5 ISA 8.3/8.4: 2D tensor, 4B elements, pad 4 dwords every 64 dwords.
__device__ __forceinline__ void tdm_load_q(const float* gaddr, unsigned lds_byte_addr) {
    unsigned long long ga = (unsigned long long)(uintptr_t)gaddr;
    v4u g0;
    g0.x = 1u;                                          // count=1, is_restore=0
    g0.y = lds_byte_addr;                               // lds_addr
    g0.z = (unsigned)(ga & 0xffffffffu);                // global_addr[31:0]
    g0.w = (unsigned)((ga >> 32) & 0x01ffffffu) | (2u << 30);  // global_addr[56:32], type=2
    v8i g1;
    g1[0] = (2 << 16)      // data_size = 2 (4 bytes)
          | (1 << 20)      // pad_enable
          | (5 << 22)      // pad_interval: 64 dwords
          | (3 << 25);     // pad_amount: 4 dwords
    g1[1] = (DDIM  << 16); // tensor_dim0 = 64      (bits 79:48, low half)
    g1[2] = (CHUNK << 16); // tensor_dim1 = 256     (bits 111:80, low half)
    g1[3] = (DDIM  << 16); // tile_dim0 = 64        (bits 127:112)
    g1[4] = CHUNK;         // tile_dim1 = 256, tile_dim2 = 0
    g1[5] = DDIM;          // tensor_dim0_stride = 64 (bits 207:160, low)
    g1[6] = 0;
    g1[7] = 0;
    v4i z4 = {0, 0, 0, 0};
    v8i z8 = {0, 0, 0, 0, 0, 0, 0, 0};
    // clang-23 / amdgpu-toolchain 6-arg form: (g0, g1, g2, g3, g4, cpol)
    __builtin_amdgcn_tensor_load_to_lds(g0, g1, z4, z4, z8, 0);
}

__global__ void zero_out_kernel(float* __restrict__ out, int n) {
    int i = (blockIdx.x * blockDim.x + threadIdx.x) * 4;
    if (i + 3 < n) {
        float4 z = {0.f, 0.f, 0.f, 0.f};
        *(float4*)(out + i) = z;
    }
}

__global__ __launch_bounds__(256)
void quadform_wmma_kernel(const float* __restrict__ q,
                          const float* __restrict__ kv,
                          float* __restrict__ out) {
    extern __shared__ float smem[];
    float* Qs = smem;                 // 256 x QSTRIDE, row-major (TDM pad mode)
    float* Sb = smem + QS_DW;         // 2 x (64 x SSTRIDE), column-major (transposed)

    const int tid   = threadIdx.x;
    const int lane  = tid & 31;
    const int wave  = tid >> 5;
    const int bid   = blockIdx.x;
    const int split = bid & (KSPLIT - 1);
    const int hn    = bid >> 2;                       // 0..127  (h*16 + n)

    const float* qg  = q  + (size_t)hn * (CHUNK * DDIM);          // 256x64 chunk
    const float* kvg = kv + (size_t)hn * (DDIM * DDIM * DDIM);    // 64x64x64 block
    const float* sbase = kvg + (size_t)split * (SLABS * DDIM * DDIM); // this split's d-range

    // Hardware LDS byte offsets of the dynamic-LDS regions
    const unsigned lds0   = __builtin_amdgcn_groupstaticsize();
    const unsigned qs_off = lds0;                       // Qs
    const unsigned sb_off = lds0 + QS_DW * 4u;          // Sb

    // ---- TDM: DMA the whole Q chunk (256x64, padded to stride 68) into LDS ----
    if (wave == 0) {
        tdm_load_q(qg, qs_off);
    }
    // ---- Async-DMA slab 0 of S, transposed into buffer 0 (b32 scatter) ----
    // global element (k, f) -> LDS dword [f*SSTRIDE + k]; lanes walk f (coalesced reads)
    #pragma unroll
    for (int j = 0; j < 16; ++j) {
        int i = tid + j * 256;                          // [0, 4096) floats
        int k = i >> 6, f = i & 63;
        async_b32_to_lds(sb_off + (unsigned)(f * SSTRIDE + k) * 4u,
                         sbase + k * DDIM + f);
    }
    if (wave == 0) {
        __builtin_amdgcn_s_wait_tensorcnt(0);
    }
    wait_asynccnt0();
    __syncthreads();

    v8f acc[2][4];
    #pragma unroll
    for (int a = 0; a < 2; ++a)
        #pragma unroll
        for (int b = 0; b < 4; ++b)
            acc[a][b] = (v8f){0.f,0.f,0.f,0.f,0.f,0.f,0.f,0.f};

    const int wrow = wave * 32;              // this wave's 32 output rows
    const int cr   = wrow + (lane & 15);     // A-fragment row (M = lane%16)
    const int hi   = lane >> 4;              // 0: K=0,1 half  1: K=2,3 half
    const int d0   = split * SLABS;
    const float scale = 1.0f / 128.0f;       // 0.5 * (1/8)^2, folded into qd

    for (int s = 0; s < SLABS; ++s) {
        const float* Scur = Sb + (s & 1) * SB_DW;

        // L2 warm-up for slab s+2
        if (s + 2 < SLABS)
            __builtin_prefetch(sbase + (size_t)(s + 2) * (DDIM * DDIM) + tid * 16, 0, 0);

        // async-DMA slab s+1 (transposed) into the other buffer; overlaps with compute.
        // Safe: all waves finished reading that buffer at the end-of-(s-1) barrier.
        if (s + 1 < SLABS) {
            const float* sn = sbase + (size_t)(s + 1) * (DDIM * DDIM);
            const unsigned boff = sb_off + (unsigned)(((s + 1) & 1) * SB_DW) * 4u;
            #pragma unroll
            for (int j = 0; j < 16; ++j) {
                int i = tid + j * 256;
                int k = i >> 6, f = i & 63;
                async_b32_to_lds(boff + (unsigned)(f * SSTRIDE + k) * 4u,
                                 sn + k * DDIM + f);
            }
        }

        const int d = d0 + s;                       // fixed d for this slab
        const float qd0 = Qs[cr * QSTRIDE + d] * scale;
        const float qd1 = Qs[(cr + 16) * QSTRIDE + d] * scale;

        #pragma unroll
        for (int ks = 0; ks < 16; ++ks) {
            const int e0 = ks * 4;                  // K-step covers e0..e0+3
            // A fragments: A'[c, d*64 + e0+j] = (scale*Q[c,d]) * Q[c,e0+j]
            float2 qe0 = *(const float2*)&Qs[cr * QSTRIDE + e0 + 2 * hi];
            float2 qe1 = *(const float2*)&Qs[(cr + 16) * QSTRIDE + e0 + 2 * hi];
            v2f a0; a0.x = qd0 * qe0.x; a0.y = qd0 * qe0.y;
            v2f a1; a1.x = qd1 * qe1.x; a1.y = qd1 * qe1.y;

            // B fragments: transposed slab -> (rows brow,brow+1 @ column f) are
            // adjacent dwords: one ds_load_b64 per fragment, bank-conflict-free.
            const int brow = e0 + 2 * hi;           // v0 row (v1 = brow+1)
            const float* bcol = Scur + (size_t)(lane & 15) * SSTRIDE + brow;
            #pragma unroll
            for (int ft = 0; ft < 4; ++ft) {
                v2f b = *(const v2f*)(bcol + ft * 16 * SSTRIDE);
                acc[0][ft] = __builtin_amdgcn_wmma_f32_16x16x4_f32(
                    false, a0, false, b, (short)0, acc[0][ft], false, false);
                acc[1][ft] = __builtin_amdgcn_wmma_f32_16x16x4_f32(
                    false, a1, false, b, (short)0, acc[1][ft], false, false);
            }
        }

        // slab s+1 must be fully resident before next iteration
        wait_asynccnt0();
        __syncthreads();
    }

    // ---- Epilogue: accumulate across K-splits (scale already folded into qd) ----
    float* og = out + (size_t)hn * (CHUNK * DDIM);
    #pragma unroll
    for (int rt = 0; rt < 2; ++rt) {
        const int crow0 = wrow + rt * 16 + hi * 8;  // C/D layout: vgpr r -> M = r + 8*hi
        #pragma unroll
        for (int ft = 0; ft < 4; ++ft) {
            const int f = ft * 16 + (lane & 15);
            #pragma unroll
            for (int r = 0; r < 8; ++r) {
                unsafeAtomicAdd(&og[(crow0 + r) * DDIM + f], acc[rt][ft][r]);
            }
        }
    }
}

extern "C" void kernel_launch(void* const* d_in, const int* in_sizes, int n_in,
                              void* d_out, int out_size, void* d_ws, size_t ws_size,
                              hipStream_t stream) {
    const float* q  = (const float*)d_in[0];   // (1,8,4096,64) f32
    const float* kv = (const float*)d_in[1];   // (1,8,16,64,64,64) f32
    float* out = (float*)d_out;                // (1,8,4096,64) f32

    // zero output (K-splits accumulate atomically)
    int zblocks = (out_size + 1023) / 1024;
    zero_out_kernel<<<zblocks, 256, 0, stream>>>(out, out_size);

    const int nblocks = 128 * KSPLIT;          // (b*h*n) * K-splits
    const size_t shmem = (QS_DW + 2 * SB_DW) * sizeof(float);  // ~104 KB (<= 320 KB/WGP)
    quadform_wmma_kernel<<<nblocks, 256, shmem, stream>>>(q, kv, out);
}